// SpatialRNN3D_10462540333427
// MI455X (gfx1250) — compile-verified
//
#include <hip/hip_runtime.h>
#include <stdint.h>

// Problem constants (match reference)
#define CCH   64          // channels per direction
#define LLEN  512         // line length (H == W == 512)
#define TILE  128         // output positions per block along the scan axis
#define HALO  33          // halo each side (32-step warm-up + 1 shift)
#define SPAN  (TILE + 2 * HALO)   // 194 staged positions

typedef uint32_t u32x4 __attribute__((ext_vector_type(4)));
typedef uint32_t u32x8 __attribute__((ext_vector_type(8)));
typedef float    f32x2 __attribute__((ext_vector_type(2)));
typedef float    f32x4 __attribute__((ext_vector_type(4)));

// ---- CDNA5 Tensor Data Mover: 2D tile (npos lines x 64 floats) -> LDS ----
// D# built per ISA 8.3/8.4: group0 = {count/type/addr}, group1 = {dims/strides}.
__device__ __forceinline__ void tdm_load_2d_to_lds(uint32_t lds_byte, uint64_t gaddr,
                                                   uint32_t npos, uint32_t pos_stride_elems) {
    u32x4 g0;
    g0[0] = 1u;                                                   // count=1, user mode
    g0[1] = lds_byte;                                             // lds_addr
    g0[2] = (uint32_t)gaddr;                                      // global_addr[31:0]
    g0[3] = ((uint32_t)(gaddr >> 32) & 0x1FFFFFFu) | (2u << 30);  // addr[56:32] | type=2

    u32x8 g1;
    g1[0] = (2u << 16);                 // workgroup_mask=0, data_size=4B, no flags
    g1[1] = ((uint32_t)CCH) << 16;      // tensor_dim0[15:0]=64  (bits 63:48)
    g1[2] = (npos & 0xFFFFu) << 16;     // tensor_dim0[31:16]=0 | tensor_dim1[15:0]=npos
    g1[3] = ((uint32_t)CCH) << 16;      // tensor_dim1[31:16]=0 | tile_dim0=64
    g1[4] = (npos & 0xFFFFu);           // tile_dim1=npos, tile_dim2=0
    g1[5] = pos_stride_elems;           // tensor_dim0_stride[31:0] (4B units)
    g1[6] = 0u;                         // stride[47:32]=0, tensor_dim1_stride[15:0]=0
    g1[7] = 0u;                         // tensor_dim1_stride[47:16]=0
    asm volatile("tensor_load_to_lds %0, %1" :: "s"(g0), "s"(g1) : "memory");
}

// One block = one wave32 = one (line, tile) pair; computes BOTH directions of
// one axis. Each lane owns 2 contiguous channels (b64 LDS reads / b64 stores).
__global__ __launch_bounds__(32) void spatial_rnn_scan(
    const float* __restrict__ x,
    const float* __restrict__ wf,   // per-channel tap, forward direction
    const float* __restrict__ wb,   // per-channel tap, backward direction
    float* __restrict__ out,
    int in_line_stride, int in_pos_stride,
    int out_line_stride, int out_pos_stride,
    int ch_f, int ch_b)
{
    __shared__ float smem[SPAN * CCH];   // 194*64*4 = 48.5 KB

    const int tid  = threadIdx.x;        // 0..31
    const int c0   = tid * 2;            // first of the 2 channels this lane owns
    const int line = blockIdx.y;
    const int q0   = blockIdx.x * TILE;
    const int pstart = q0 - HALO;        // global position of staged slot 0

    // Clamped in-line window actually present in memory
    const int lo_p = (pstart < 0) ? 0 : pstart;
    const int hi_p = (pstart + SPAN > LLEN) ? LLEN : (pstart + SPAN);
    const int lo_slot = lo_p - pstart;   // 0 or HALO at line start
    const int hi_slot = hi_p - pstart;   // SPAN or less at line end

    const uint32_t smem_base = (uint32_t)(uintptr_t)(&smem[0]);

    // ---- TDM first: the DMA engine streams HBM->LDS while this wave works ----
    {
        const uint64_t gaddr = (uint64_t)(uintptr_t)
            (x + (size_t)line * in_line_stride + (size_t)lo_p * in_pos_stride);
        tdm_load_2d_to_lds(smem_base + (uint32_t)(lo_slot * CCH * 4), gaddr,
                           (uint32_t)(hi_p - lo_p), (uint32_t)in_pos_stride);
    }

    // Hoist weight loads so they overlap the TDM transfer.
    const float wfx = wf[c0], wfy = wf[c0 + 1];
    const float wbx = wb[c0], wby = wb[c0 + 1];

    // ---- Zero-fill out-of-line halo slots (disjoint from the TDM region) ----
    {
        const f32x4 z = {0.f, 0.f, 0.f, 0.f};
        f32x4* head = (f32x4*)smem;
        for (int i = tid; i < lo_slot * (CCH / 4); i += 32) head[i] = z;
        f32x4* tail = (f32x4*)&smem[hi_slot * CCH];
        for (int i = tid; i < (SPAN - hi_slot) * (CCH / 4); i += 32) tail[i] = z;
    }

    __builtin_amdgcn_s_wait_tensorcnt(0);   // TDM data landed in LDS
    __syncthreads();                        // order DS zero-fills vs. reads

    const f32x2* sm2 = (const f32x2*)smem;     // index = slot*32 + tid
    const int ostep2 = out_pos_stride / 2;     // store step in f32x2 units

    // -------- Forward (out[p] <- in[p-1]) : S = relu(w*x[p-1]) + wp*S --------
    {
        const float px = fmaxf(wfx, 0.f), py = fmaxf(wfy, 0.f);
        int qstart = q0 - (HALO - 1); if (qstart < 0) qstart = 0;
        int slot = qstart - pstart;                       // >= 1
        f32x2 xp = sm2[(slot - 1) * 32 + tid];            // zero if out-of-line
        float Sx = 0.f, Sy = 0.f;
        for (int q = qstart; q < q0; ++q, ++slot) {       // warm-up, no stores
            f32x2 xc = sm2[slot * 32 + tid];
            Sx = fmaf(px, Sx, fmaxf(wfx * xp.x, 0.f));
            Sy = fmaf(py, Sy, fmaxf(wfy * xp.y, 0.f));
            xp = xc;
        }
        f32x2* op = (f32x2*)(out + (size_t)line * out_line_stride
                                 + (size_t)q0 * out_pos_stride + ch_f + c0);
        #pragma unroll 4
        for (int k = 0; k < TILE; ++k, ++slot) {          // main, always store
            f32x2 xc = sm2[slot * 32 + tid];
            Sx = fmaf(px, Sx, fmaxf(wfx * xp.x, 0.f));
            Sy = fmaf(py, Sy, fmaxf(wfy * xp.y, 0.f));
            f32x2 r; r.x = xc.x + Sx; r.y = xc.y + Sy;
            __builtin_nontemporal_store(r, op);
            op += ostep2;
            xp = xc;
        }
    }

    // -------- Backward (out[p] <- in[p+1]) --------
    {
        const float px = fmaxf(wbx, 0.f), py = fmaxf(wby, 0.f);
        const int qlast = q0 + TILE - 1;
        int qstart = qlast + (HALO - 1); if (qstart > LLEN - 1) qstart = LLEN - 1;
        int slot = qstart - pstart;                       // <= SPAN-2
        f32x2 xn = sm2[(slot + 1) * 32 + tid];            // zero if out-of-line
        float Sx = 0.f, Sy = 0.f;
        for (int q = qstart; q > qlast; --q, --slot) {    // warm-up, no stores
            f32x2 xc = sm2[slot * 32 + tid];
            Sx = fmaf(px, Sx, fmaxf(wbx * xn.x, 0.f));
            Sy = fmaf(py, Sy, fmaxf(wby * xn.y, 0.f));
            xn = xc;
        }
        f32x2* op = (f32x2*)(out + (size_t)line * out_line_stride
                                 + (size_t)qlast * out_pos_stride + ch_b + c0);
        #pragma unroll 4
        for (int k = 0; k < TILE; ++k, --slot) {          // main, always store
            f32x2 xc = sm2[slot * 32 + tid];
            Sx = fmaf(px, Sx, fmaxf(wbx * xn.x, 0.f));
            Sy = fmaf(py, Sy, fmaxf(wby * xn.y, 0.f));
            f32x2 r; r.x = xc.x + Sx; r.y = xc.y + Sy;
            __builtin_nontemporal_store(r, op);
            op -= ostep2;
            xn = xc;
        }
    }
}

extern "C" void kernel_launch(void* const* d_in, const int* in_sizes, int n_in,
                              void* d_out, int out_size, void* d_ws, size_t ws_size,
                              hipStream_t stream) {
    (void)in_sizes; (void)n_in; (void)out_size; (void)d_ws; (void)ws_size;

    const float* x  = (const float*)d_in[0];
    const float* kr = (const float*)d_in[1];  // (1,3,C,1) active tap [0][0][c][0] -> +0
    const float* kl = (const float*)d_in[2];  // (1,3,C,1) active tap [0][2][c][0] -> +2C
    const float* kd = (const float*)d_in[3];  // (3,1,C,1) active tap [0][0][c][0] -> +0
    const float* ku = (const float*)d_in[4];  // (3,1,C,1) active tap [2][0][c][0] -> +2C
    float* out = (float*)d_out;

    dim3 grid(LLEN / TILE, LLEN);  // (tiles, lines)
    dim3 blk(32);                  // one wave32 per block

    // Horizontal axis: lines are rows. right -> ch[0:64), left -> ch[64:128).
    spatial_rnn_scan<<<grid, blk, 0, stream>>>(
        x, kr + 0, kl + 2 * CCH, out,
        /*in_line_stride=*/LLEN * CCH, /*in_pos_stride=*/CCH,
        /*out_line_stride=*/LLEN * 4 * CCH, /*out_pos_stride=*/4 * CCH,
        /*ch_f=*/0, /*ch_b=*/CCH);

    // Vertical axis: lines are columns. down -> ch[128:192), up -> ch[192:256).
    spatial_rnn_scan<<<grid, blk, 0, stream>>>(
        x, kd + 0, ku + 2 * CCH, out,
        /*in_line_stride=*/CCH, /*in_pos_stride=*/LLEN * CCH,
        /*out_line_stride=*/4 * CCH, /*out_pos_stride=*/LLEN * 4 * CCH,
        /*ch_f=*/2 * CCH, /*ch_b=*/3 * CCH);
}